// MeanMessageAggregator_72052371357814
// MI455X (gfx1250) — compile-verified
//
#include <hip/hip_runtime.h>
#include <hip/hip_bf16.h>

typedef __attribute__((ext_vector_type(2))) float v2f;
typedef __attribute__((ext_vector_type(8))) float v8f;

// ---------------------------------------------------------------------------
// Kernel 1: segment boundaries via binary search over sorted node_ids.
// seg_start[v] = lower_bound(node_ids, v), for v in [0, V]; seg_start[V] = N.
// ---------------------------------------------------------------------------
__global__ void seg_bounds_kernel(const int* __restrict__ ids, int N, int V,
                                  int* __restrict__ seg_start) {
    int v = blockIdx.x * blockDim.x + threadIdx.x;
    if (v > V) return;
    int lo = 0, hi = N;
    while (lo < hi) {
        int mid = (lo + hi) >> 1;
        if (ids[mid] < v) lo = mid + 1; else hi = mid;
    }
    seg_start[v] = lo;
}

// ---------------------------------------------------------------------------
// Kernel 2: windowed segment-mean as f32 WMMA SpMM.
// One block = 16 consecutive nodes, 512 threads = 16 wave32 = 16 column tiles
// of 16 (covers D=256). Sorted ids => the 16 windows occupy one contiguous
// message-row range; sweep it in K=4 chunks with V_WMMA_F32_16X16X4_F32:
//   A[m,k] = 1.0 if row (base+k) is inside node m's last-n window, else 0
//   B[k,n] = messages[base+k, colbase+n]
//   C += A x B   (exact f32 accumulation on the matrix pipe)
// A layout (32-bit A 16x4): lane M = lane&15; lanes 0-15 hold K={0,1},
// lanes 16-31 hold K={2,3}. B mirrors with N = lane&15.
// C/D layout: VGPR j -> M = j + 8*(lane>=16), N = lane&15.
// Window membership tested as (unsigned)(row - ws) < wlen (interval trick).
// ---------------------------------------------------------------------------
__global__ __launch_bounds__(512) void seg_mean_wmma_kernel(
    const float* __restrict__ messages, const float* __restrict__ timestamps,
    const int* __restrict__ seg_start, const int* __restrict__ nn_ptr,
    float* __restrict__ out_has, float* __restrict__ out_msg,
    float* __restrict__ out_ts, int N, int V, int D) {

    __shared__ int   s_s[17];     // segment boundaries for nodes v0..v0+16
    __shared__ int   s_ws[16];    // window start per node
    __shared__ float s_rden[16];  // 1.0f / max(kept_count, 1) per node

    const int v0 = blockIdx.x * 16;
    const int t  = threadIdx.x;
    const int nn = nn_ptr[0];

    if (t < 17) {
        int v = v0 + t;
        s_s[t] = seg_start[v <= V ? v : V];   // clamp for tail blocks
    }
    __syncthreads();
    if (t < 16) {
        int s = s_s[t], e = s_s[t + 1];
        int ws = e - nn; if (ws < s) ws = s;  // window = last <= nn rows
        s_ws[t] = ws;
        int kept = e - ws; if (kept < 1) kept = 1;
        s_rden[t] = 1.0f / (float)kept;       // one exact divide per node
    }
    __syncthreads();

    const int lane   = t & 31;
    const int wave   = t >> 5;
    const int nwaves = blockDim.x >> 5;
    const int mrow   = lane & 15;             // M for A / N for B
    const int khalf  = (lane >> 4) << 1;      // lanes 0-15: K 0,1; 16-31: K 2,3
    const int ws_m   = s_ws[mrow];
    const unsigned wlen = (unsigned)(s_s[mrow + 1] - ws_m);  // window length
    const int mhalf  = (lane >> 4) * 8;       // C/D half select

    const int rowlo = s_ws[0];                // windows sorted; min start
    const int rowhi = s_s[16];                // end of last segment

    // per-lane reciprocal denominators for its 8 C/D rows (outside hot loop)
    float rden[8];
    #pragma unroll
    for (int j = 0; j < 8; ++j) rden[j] = s_rden[mhalf + j];

    const bool fullblock = (v0 + 16 <= V);    // uniform: scalar branch

    for (int colbase = wave * 16; colbase < D; colbase += nwaves * 16) {
        const int col = colbase + mrow;
        v8f acc0 = {0.f, 0.f, 0.f, 0.f, 0.f, 0.f, 0.f, 0.f};
        v8f acc1 = {0.f, 0.f, 0.f, 0.f, 0.f, 0.f, 0.f, 0.f};

        // 32-bit element offset off a uniform base -> saddr+voffset+imm loads
        int off = (rowlo + khalf) * D + col;
        int rr  = rowlo + khalf - ws_m;       // row index relative to window
        int base = rowlo;

        // unrolled: 2 chunks (8 rows) per iteration, 4 loads in one clause,
        // two independent WMMA accumulation chains
        for (; base + 8 <= rowhi; base += 8) {
            v2f a0, a1, b0, b1;
            a0.x = ((unsigned)(rr    ) < wlen) ? 1.0f : 0.0f;
            a0.y = ((unsigned)(rr + 1) < wlen) ? 1.0f : 0.0f;
            a1.x = ((unsigned)(rr + 4) < wlen) ? 1.0f : 0.0f;
            a1.y = ((unsigned)(rr + 5) < wlen) ? 1.0f : 0.0f;
            b0.x = messages[off];
            b0.y = messages[off + D];
            b1.x = messages[off + 4 * D];
            b1.y = messages[off + 5 * D];
            acc0 = __builtin_amdgcn_wmma_f32_16x16x4_f32(
                false, a0, false, b0, (short)0, acc0, false, false);
            acc1 = __builtin_amdgcn_wmma_f32_16x16x4_f32(
                false, a1, false, b1, (short)0, acc1, false, false);
            rr  += 8;
            off += 8 * D;
        }

        // remainder: up to 2 chunks, clamped row reads (rows < rowhi <= N valid)
        for (; base < rowhi; base += 4) {
            const int r0 = base + khalf;
            const int r1 = r0 + 1;
            v2f a, b;
            a.x = ((unsigned)(rr    ) < wlen) ? 1.0f : 0.0f;
            a.y = ((unsigned)(rr + 1) < wlen) ? 1.0f : 0.0f;
            const int c0 = r0 < rowhi - 1 ? r0 : rowhi - 1;
            const int c1 = r1 < rowhi - 1 ? r1 : rowhi - 1;
            b.x = messages[c0 * D + col];
            b.y = messages[c1 * D + col];
            acc0 = __builtin_amdgcn_wmma_f32_16x16x4_f32(
                false, a, false, b, (short)0, acc0, false, false);
            rr += 4;
        }

        // combine chains, scale by 1/kept, store mean tile (imm-offset stores)
        const int oo = (v0 + mhalf) * D + col;
        if (fullblock) {
            #pragma unroll
            for (int j = 0; j < 8; ++j)
                out_msg[oo + j * D] = (acc0[j] + acc1[j]) * rden[j];
        } else {
            #pragma unroll
            for (int j = 0; j < 8; ++j)
                if (v0 + mhalf + j < V)
                    out_msg[oo + j * D] = (acc0[j] + acc1[j]) * rden[j];
        }
    }

    // has_msg flag + last timestamp (one thread per node)
    if (t < 16) {
        const int v = v0 + t;
        if (v < V) {
            const int s = s_s[t], e = s_s[t + 1];
            const bool has = e > s;
            out_has[v] = has ? 1.0f : 0.0f;
            out_ts[v]  = has ? timestamps[e - 1] : 0.0f;
        }
    }
}

extern "C" void kernel_launch(void* const* d_in, const int* in_sizes, int n_in,
                              void* d_out, int out_size, void* d_ws, size_t ws_size,
                              hipStream_t stream) {
    const int*   node_ids   = (const int*)d_in[0];
    const float* messages   = (const float*)d_in[1];
    const float* timestamps = (const float*)d_in[2];
    // d_in[3] = memory (unused by the mean aggregator)
    const int*   nn_ptr     = (const int*)d_in[4];

    const int N = in_sizes[0];
    const int D = in_sizes[1] / N;            // 256
    const int V = out_size / (D + 2);         // has[V] + msg[V*D] + ts[V]

    int* seg_start = (int*)d_ws;              // V+1 ints of scratch

    float* out     = (float*)d_out;
    float* out_has = out;
    float* out_msg = out + V;
    float* out_ts  = out + V + (long)V * D;

    seg_bounds_kernel<<<(V + 1 + 255) / 256, 256, 0, stream>>>(
        node_ids, N, V, seg_start);

    seg_mean_wmma_kernel<<<(V + 15) / 16, 512, 0, stream>>>(
        messages, timestamps, seg_start, nn_ptr,
        out_has, out_msg, out_ts, N, V, D);
}